// SEM_43662637531231
// MI455X (gfx1250) — compile-verified
//
#include <hip/hip_runtime.h>
#include <hip/hip_bf16.h>
#include <math.h>

typedef __attribute__((ext_vector_type(2))) float v2f;
typedef __attribute__((ext_vector_type(8))) float v8f;

constexpr long NSP  = 102400;  // 320*320 spatial positions
constexpr int  HGT  = 320;
constexpr int  WIMG = 320;
constexpr int  CDIM = 64;

static __device__ __forceinline__ v8f wmma_f32_4(v2f a, v2f b, v8f c) {
  // D(16x16,f32) = A(16x4,f32) x B(4x16,f32) + C
  return __builtin_amdgcn_wmma_f32_16x16x4_f32(false, a, false, b, (short)0, c,
                                               false, false);
}

// ---------------------------------------------------------------------------
// 1) Channel LayerNorm for both inputs. One thread per (batch, spatial pos).
// ---------------------------------------------------------------------------
__global__ void ln2_kernel(const float* __restrict__ R, const float* __restrict__ S,
                           const float* __restrict__ w1, const float* __restrict__ b1,
                           const float* __restrict__ w2, const float* __restrict__ b2,
                           float* __restrict__ xR, float* __restrict__ xp) {
  long idx = (long)blockIdx.x * blockDim.x + threadIdx.x;
  if (idx >= 2 * NSP) return;
  long b = idx / NSP, n = idx % NSP;
  const float* Rb = R + b * CDIM * NSP + n;
  const float* Sb = S + b * CDIM * NSP + n;
  float* xRb = xR + b * CDIM * NSP + n;
  float* xpb = xp + b * CDIM * NSP + n;
  float s1 = 0.f, q1 = 0.f, s2 = 0.f, q2 = 0.f;
  for (int c = 0; c < CDIM; ++c) {
    float r = Rb[(long)c * NSP]; s1 += r; q1 += r * r;
    float s = Sb[(long)c * NSP]; s2 += s; q2 += s * s;
  }
  float mu1 = s1 / CDIM, var1 = q1 / CDIM - mu1 * mu1;
  float mu2 = s2 / CDIM, var2 = q2 / CDIM - mu2 * mu2;
  float is1 = rsqrtf(var1 + 1e-5f), is2 = rsqrtf(var2 + 1e-5f);
  for (int c = 0; c < CDIM; ++c) {
    xRb[(long)c * NSP] = (Rb[(long)c * NSP] - mu1) * is1 * w1[c] + b1[c];
    xpb[(long)c * NSP] = (Sb[(long)c * NSP] - mu2) * is2 * w2[c] + b2[c];
  }
}

// ---------------------------------------------------------------------------
// 2) conv1x1 as M-blocked WMMA f32 GEMM:
//       Y[o,n] = sum_c W[o,c] * (X0[c,n] (+ X1[c,n]))
//    Each wave owns one 16-wide spatial tile and ALL Cout channels
//    (MT = Cout/16 accumulator tiles), so every X element is loaded from
//    global exactly once per conv.  Weights (Cout x 64, <=32KB) are staged
//    in LDS once per workgroup and feed the A operand via ds loads.
//    Compile-time flags: fused second input (x_mid = xp + xR), bias, ReLU.
// ---------------------------------------------------------------------------
template <int MT, bool HASX1, bool HASBIAS, bool RELU>
__global__ void conv1x1_wmma(const float* __restrict__ W, const float* __restrict__ X0,
                             const float* __restrict__ X1, const float* __restrict__ bias,
                             float* __restrict__ Y) {
  constexpr int CIN  = 64;
  constexpr int COUT = MT * 16;
  __shared__ float wl[COUT * CIN];
  for (int i = threadIdx.x; i < COUT * CIN; i += blockDim.x) wl[i] = W[i];
  __syncthreads();

  const int lane = threadIdx.x & 31;
  const int wv   = threadIdx.x >> 5;
  const int half = lane >> 4;          // lane group 0/1 -> K {0,1} vs {2,3}
  const int l16  = lane & 15;
  const int tn   = blockIdx.x * (blockDim.x >> 5) + wv;
  const long n0  = (long)tn * 16;
  const long b   = blockIdx.y;
  const float* Xb0 = X0 + b * (long)CIN * NSP;
  const float* Xb1 = HASX1 ? (X1 + b * (long)CIN * NSP) : nullptr;
  float* Yb = Y + b * (long)COUT * NSP;

  v8f zero = {};
  v8f acc[MT];
#pragma unroll
  for (int m = 0; m < MT; ++m) acc[m] = zero;

  for (int k0 = 0; k0 < CIN; k0 += 4) {
    const int kk = k0 + 2 * half;
    const float* xc = Xb0 + (long)kk * NSP + n0 + l16;
    v2f bb;                                   // B[K=kk,kk+1][N=l16]
    bb.x = xc[0];
    bb.y = xc[NSP];
    if constexpr (HASX1) {                    // fused x_mid = xp + xR
      const float* xc1 = Xb1 + (long)kk * NSP + n0 + l16;
      bb.x += xc1[0];
      bb.y += xc1[NSP];
    }
#pragma unroll
    for (int m = 0; m < MT; ++m) {
      v2f a;                                  // A[M=l16][K=kk,kk+1] from LDS
      a.x = wl[(m * 16 + l16) * CIN + kk];
      a.y = wl[(m * 16 + l16) * CIN + kk + 1];
      acc[m] = wmma_f32_4(a, bb, acc[m]);
    }
  }
#pragma unroll
  for (int m = 0; m < MT; ++m) {
#pragma unroll
    for (int r = 0; r < 8; ++r) {
      int mm = m * 16 + r + 8 * half;
      float v = acc[m][r];
      if constexpr (HASBIAS) v += bias[mm];
      if constexpr (RELU) v = fmaxf(v, 0.0f);
      Yb[(long)mm * NSP + n0 + l16] = v;
    }
  }
}

// ---------------------------------------------------------------------------
// 3) Grouped 3x3 conv (groups=64, 2-in/2-out per group), SAME pad.
//    One thread computes BOTH output channels of a group -> each input
//    neighborhood is loaded once and reused for 2 outputs.
// ---------------------------------------------------------------------------
__global__ void dwconv3x3_kernel(const float* __restrict__ X, const float* __restrict__ Wt,
                                 float* __restrict__ Y) {
  long idx = (long)blockIdx.x * blockDim.x + threadIdx.x;  // over 2*64*NSP groups
  const long total = 2L * 64 * NSP;
  if (idx >= total) return;
  long n = idx % NSP;
  int g  = (int)((idx / NSP) % 64);
  int b  = (int)(idx / (NSP * 64));
  int y = (int)(n / WIMG), x = (int)(n % WIMG);
  float acc0 = 0.f, acc1 = 0.f;
#pragma unroll
  for (int i = 0; i < 2; ++i) {
    const float* Xc  = X + (long)(b * 128 + g * 2 + i) * NSP;
    const float* wp0 = Wt + ((g * 2 + 0) * 2 + i) * 9;
    const float* wp1 = Wt + ((g * 2 + 1) * 2 + i) * 9;
#pragma unroll
    for (int dy = 0; dy < 3; ++dy) {
      int yy = y + dy - 1;
      if (yy < 0 || yy >= HGT) continue;
#pragma unroll
      for (int dx = 0; dx < 3; ++dx) {
        int xx = x + dx - 1;
        if (xx < 0 || xx >= WIMG) continue;
        float v = Xc[(long)yy * WIMG + xx];
        acc0 += v * wp0[dy * 3 + dx];
        acc1 += v * wp1[dy * 3 + dx];
      }
    }
  }
  Y[(long)(b * 128 + g * 2 + 0) * NSP + n] = acc0;
  Y[(long)(b * 128 + g * 2 + 1) * NSP + n] = acc1;
}

// ---------------------------------------------------------------------------
// 4) Reciprocal L2 norms over N for q1,q2,kx,ky. One block per (sel,b,chan).
//    rn layout: [sel(4)][b(2)][c(64)]
// ---------------------------------------------------------------------------
__global__ void rnorm_kernel(const float* __restrict__ qq, const float* __restrict__ kvx,
                             const float* __restrict__ kvy, float* __restrict__ rn) {
  __shared__ float red[256];
  int sel = blockIdx.x >> 7;
  int rem = blockIdx.x & 127;
  int b = rem >> 6;
  int c = rem & 63;
  const float* src;
  if (sel == 0)      src = qq  + (long)(b * 128 + c) * NSP;        // q1
  else if (sel == 1) src = qq  + (long)(b * 128 + 64 + c) * NSP;   // q2
  else if (sel == 2) src = kvx + (long)(b * 128 + c) * NSP;        // kx
  else               src = kvy + (long)(b * 128 + c) * NSP;        // ky
  float s = 0.f;
  for (long n = threadIdx.x; n < NSP; n += blockDim.x) {
    float v = src[n];
    s += v * v;
  }
  red[threadIdx.x] = s;
  __syncthreads();
  for (int st = 128; st > 0; st >>= 1) {
    if ((int)threadIdx.x < st) red[threadIdx.x] += red[threadIdx.x + st];
    __syncthreads();
  }
  if (threadIdx.x == 0)
    rn[sel * 128 + b * 64 + c] = 1.0f / fmaxf(sqrtf(red[0]), 1e-12f);
}

__global__ void zero_kernel(float* __restrict__ p, int n) {
  int i = blockIdx.x * blockDim.x + threadIdx.x;
  if (i < n) p[i] = 0.f;
}

// ---------------------------------------------------------------------------
// 5) Attention logits via WMMA: logits[b,h] += q1n(16xK) * kyn(Kx16)
//                                           + q2n(16xK) * kxn(Kx16),
//    K chunked over N=102400 across 64 waves per (b,h), atomic fp32 combine.
// ---------------------------------------------------------------------------
__global__ void attn_logits_kernel(const float* __restrict__ qq, const float* __restrict__ kvx,
                                   const float* __restrict__ kvy, const float* __restrict__ rn,
                                   float* __restrict__ logits) {
  const int lane = threadIdx.x & 31;
  const int wv   = threadIdx.x >> 5;
  const int half = lane >> 4;
  const int l16  = lane & 15;
  const int chunk = blockIdx.x * (blockDim.x >> 5) + wv;   // 0..63
  const int h = blockIdx.y;
  const int b = blockIdx.z;
  const long CH = NSP / 64;                                 // 1600 positions
  const long n0 = (long)chunk * CH;
  const int ch = h * 16 + l16;           // row channel (A) == col channel (B)
  const float rq1 = rn[0 * 128 + b * 64 + ch];
  const float rq2 = rn[1 * 128 + b * 64 + ch];
  const float rkx = rn[2 * 128 + b * 64 + ch];
  const float rky = rn[3 * 128 + b * 64 + ch];
  const float* q1p = qq  + (long)(b * 128 + ch) * NSP;
  const float* q2p = qq  + (long)(b * 128 + 64 + ch) * NSP;
  const float* kxp = kvx + (long)(b * 128 + ch) * NSP;
  const float* kyp = kvy + (long)(b * 128 + ch) * NSP;

  v8f acc = {};
  for (long s = 0; s < CH; s += 4) {
    long n = n0 + s + 2 * half;
    v2f a1; a1.x = q1p[n] * rq1; a1.y = q1p[n + 1] * rq1;
    v2f b1; b1.x = kyp[n] * rky; b1.y = kyp[n + 1] * rky;
    acc = wmma_f32_4(a1, b1, acc);                  // q1 . ky^T
    v2f a2; a2.x = q2p[n] * rq2; a2.y = q2p[n + 1] * rq2;
    v2f b2; b2.x = kxp[n] * rkx; b2.y = kxp[n + 1] * rkx;
    acc = wmma_f32_4(a2, b2, acc);                  // + q2 . kx^T
  }
  float* Lrow = logits + (long)(b * 4 + h) * 256;
#pragma unroll
  for (int r = 0; r < 8; ++r) {
    int c = r + 8 * half;
    atomicAdd(&Lrow[c * 16 + l16], acc[r]);
  }
}

// ---------------------------------------------------------------------------
// 6) softmax(logits * temperature) over d; one thread per (b,h,c) row.
// ---------------------------------------------------------------------------
__global__ void softmax_kernel(const float* __restrict__ logits,
                               const float* __restrict__ temperature,
                               float* __restrict__ attn) {
  int row = threadIdx.x;
  if (row >= 128) return;
  int h = (row >> 4) & 3;
  float t = temperature[h];
  const float* L = logits + row * 16;
  float v[16];
  float m = -1e30f;
#pragma unroll
  for (int d = 0; d < 16; ++d) { v[d] = L[d] * t; m = fmaxf(m, v[d]); }
  float s = 0.f;
#pragma unroll
  for (int d = 0; d < 16; ++d) { v[d] = __expf(v[d] - m); s += v[d]; }
  float inv = 1.0f / s;
  float* A = attn + row * 16;
#pragma unroll
  for (int d = 0; d < 16; ++d) A[d] = v[d] * inv;
}

// ---------------------------------------------------------------------------
// 7) out = gamma1 * attn@vx + gamma2 * attn@vy  via WMMA (K=16, 4 steps),
//    dual accumulators, gamma epilogue.
// ---------------------------------------------------------------------------
__global__ void apply_attn_kernel(const float* __restrict__ attn, const float* __restrict__ kvx,
                                  const float* __restrict__ kvy, const float* __restrict__ g1,
                                  const float* __restrict__ g2, float* __restrict__ out) {
  const int lane = threadIdx.x & 31;
  const int wv   = threadIdx.x >> 5;
  const int half = lane >> 4;
  const int l16  = lane & 15;
  const int tn   = blockIdx.x * (blockDim.x >> 5) + wv;
  const long n0  = (long)tn * 16;
  const int h = blockIdx.y;
  const int b = blockIdx.z;
  const float* A = attn + (long)(b * 4 + h) * 256;   // [c][d] row-major

  v8f accx = {}, accy = {};
#pragma unroll
  for (int k0 = 0; k0 < 16; k0 += 4) {
    int kk = k0 + 2 * half;
    v2f a; a.x = A[l16 * 16 + kk]; a.y = A[l16 * 16 + kk + 1];
    const float* vx = kvx + (long)(b * 128 + 64 + h * 16 + kk) * NSP + n0 + l16;
    v2f bx; bx.x = vx[0]; bx.y = vx[NSP];
    accx = wmma_f32_4(a, bx, accx);
    const float* vy = kvy + (long)(b * 128 + 64 + h * 16 + kk) * NSP + n0 + l16;
    v2f by; by.x = vy[0]; by.y = vy[NSP];
    accy = wmma_f32_4(a, by, accy);
  }
  float G1 = g1[0], G2 = g2[0];
#pragma unroll
  for (int r = 0; r < 8; ++r) {
    int c = r + 8 * half;
    out[(long)(b * 64 + h * 16 + c) * NSP + n0 + l16] = G1 * accx[r] + G2 * accy[r];
  }
}

// ---------------------------------------------------------------------------
extern "C" void kernel_launch(void* const* d_in, const int* in_sizes, int n_in,
                              void* d_out, int out_size, void* d_ws, size_t ws_size,
                              hipStream_t stream) {
  (void)in_sizes; (void)n_in; (void)out_size; (void)ws_size;
  const float* input_R     = (const float*)d_in[0];
  const float* input_S     = (const float*)d_in[1];
  const float* ln1_w       = (const float*)d_in[2];
  const float* ln1_b       = (const float*)d_in[3];
  const float* ln2_w       = (const float*)d_in[4];
  const float* ln2_b       = (const float*)d_in[5];
  const float* kv_w        = (const float*)d_in[6];
  const float* kv_dw_w     = (const float*)d_in[7];
  const float* q_w         = (const float*)d_in[8];
  const float* q_dw_w      = (const float*)d_in[9];
  const float* ky_w        = (const float*)d_in[10];
  const float* ky_dw_w     = (const float*)d_in[11];
  const float* temperature = (const float*)d_in[12];
  const float* gamma1      = (const float*)d_in[13];
  const float* gamma2      = (const float*)d_in[14];
  const float* mlp_w1      = (const float*)d_in[15];
  const float* mlp_b1      = (const float*)d_in[16];
  const float* mlp_w2      = (const float*)d_in[17];
  const float* mlp_b2      = (const float*)d_in[18];

  float* ws = (float*)d_ws;
  const long szC64  = 2L * 64 * NSP;
  const long szC128 = 2L * 128 * NSP;
  float* xR     = ws;                 // [2,64,N]
  float* xp     = xR + szC64;         // [2,64,N]
  float* pre    = xp + szC64;         // [2,128,N] conv1x1 out (pre-dwconv); later aliased as `out`
  float* qq     = pre + szC128;       // [2,128,N]
  float* kvx    = qq + szC128;        // [2,128,N]
  float* kvy    = kvx + szC128;       // [2,128,N]
  float* rn     = kvy + szC128;       // [4][2][64] reciprocal norms
  float* logits = rn + 512;           // [2,4,16,16]
  float* attn   = logits + 2048;      // [2,4,16,16]
  float* outb   = pre;                // reuse: pre dead after last dwconv
  float* hmid   = xR;                 // reuse: xR dead after ky conv

  const dim3 blk(128);                               // 4 waves/block
  const dim3 gGemm((unsigned)(NSP / 16 / 4), 2);     // n-tiles x batch
  const unsigned gDw = (unsigned)(2L * 64 * NSP / 256);

  // LayerNorms
  ln2_kernel<<<(unsigned)((2 * NSP + 255) / 256), 256, 0, stream>>>(
      input_R, input_S, ln1_w, ln1_b, ln2_w, ln2_b, xR, xp);

  // qq = dw3x3(conv1x1(xp + xR, kv_w))
  conv1x1_wmma<8, true, false, false><<<gGemm, blk, 0, stream>>>(kv_w, xp, xR, nullptr, pre);
  dwconv3x3_kernel<<<gDw, 256, 0, stream>>>(pre, kv_dw_w, qq);
  // kvx = dw3x3(conv1x1(xp, q_w))
  conv1x1_wmma<8, false, false, false><<<gGemm, blk, 0, stream>>>(q_w, xp, nullptr, nullptr, pre);
  dwconv3x3_kernel<<<gDw, 256, 0, stream>>>(pre, q_dw_w, kvx);
  // kvy = dw3x3(conv1x1(xR, ky_w))
  conv1x1_wmma<8, false, false, false><<<gGemm, blk, 0, stream>>>(ky_w, xR, nullptr, nullptr, pre);
  dwconv3x3_kernel<<<gDw, 256, 0, stream>>>(pre, ky_dw_w, kvy);

  // L2 norms, attention logits (atomic WMMA accumulation), softmax
  rnorm_kernel<<<512, 256, 0, stream>>>(qq, kvx, kvy, rn);
  zero_kernel<<<8, 256, 0, stream>>>(logits, 2048);
  attn_logits_kernel<<<dim3(16, 4, 2), blk, 0, stream>>>(qq, kvx, kvy, rn, logits);
  softmax_kernel<<<1, 128, 0, stream>>>(logits, temperature, attn);

  // out = g1*attn@vx + g2*attn@vy
  apply_attn_kernel<<<dim3((unsigned)(NSP / 16 / 4), 4, 2), blk, 0, stream>>>(
      attn, kvx, kvy, gamma1, gamma2, outb);

  // MLP: relu(W1@out + b1) -> W2@hmid + b2
  conv1x1_wmma<4, false, true, true><<<gGemm, blk, 0, stream>>>(mlp_w1, outb, nullptr, mlp_b1, hmid);
  conv1x1_wmma<4, false, true, false><<<gGemm, blk, 0, stream>>>(mlp_w2, hmid, nullptr, mlp_b2,
                                                                 (float*)d_out);
}